// StructureModule_75067438399490
// MI455X (gfx1250) — compile-verified
//
#include <hip/hip_runtime.h>
#include <hip/hip_bf16.h>
#include <math.h>

// Problem constants (B=1)
#define NN   512
#define CSD  384
#define CZD  128
#define HD   12
#define CD   16
#define PD   4
#define PVD  8
#define NBD  64
#define HC   (HD*CD)       // 192
#define HP3  (HD*PD*3)     // 144
#define HPV3 (HD*PVD*3)    // 288
#define FEAT (HD*(CD+PVD*4+CZD)) // 2112

typedef float v2f __attribute__((ext_vector_type(2)));
typedef float v8f __attribute__((ext_vector_type(8)));

__device__ __forceinline__ v8f wmma4(v2f a, v2f b, v8f c) {
  // D = A(16x4,f32) * B(4x16,f32) + C(16x16,f32)
  return __builtin_amdgcn_wmma_f32_16x16x4_f32(false, a, false, b, (short)0, c,
                                               false, false);
}

// ---------------------------------------------------------------------------
// Transpose + zero-pad: Wt[c][k] = (c < Nc) ? W[k][c] : 0.   Wt is [NcPad][K].
// ---------------------------------------------------------------------------
__global__ void transpose_pad_kernel(const float* __restrict__ W, float* __restrict__ Wt,
                                     int K, int Nc, int NcPad) {
  int idx = blockIdx.x * blockDim.x + threadIdx.x;
  if (idx >= K * NcPad) return;
  int c = idx / K, k = idx % K;
  Wt[idx] = (c < Nc) ? W[(size_t)k * Nc + c] : 0.0f;
}

// ---------------------------------------------------------------------------
// C[M x Nc] = A[M x K] @ B + bias with B pre-transposed/padded as Bt[NcPad][K].
// Branch-free inner loop: two b64 loads + one WMMA per K-step.
// grid.x = M/16, grid.y = NcPad/16, block = 32 (one wave / 16x16 tile).
// ---------------------------------------------------------------------------
template <bool GUARD>
__global__ __launch_bounds__(32)
void gemm_bt_kernel(const float* __restrict__ A, const float* __restrict__ Bt,
                    const float* __restrict__ bias, float* __restrict__ out,
                    int K, int Nc) {
  int lane = threadIdx.x;
  int lo = lane & 15, hi = lane >> 4;
  int r0 = blockIdx.x * 16;
  int col = blockIdx.y * 16 + lo;
  const float* Ap = A  + (size_t)(r0 + lo) * K + 2 * hi;   // A row = lo
  const float* Bp = Bt + (size_t)col * K + 2 * hi;         // padded: in-bounds
  v8f acc = {};
#pragma unroll 4
  for (int k = 0; k < K; k += 4) {
    __builtin_prefetch(Ap + 64, 0, 0);       // stream A (z) toward L2
    v2f a = *(const v2f*)Ap;
    v2f b = *(const v2f*)Bp;
    acc = wmma4(a, b, acc);
    Ap += 4; Bp += 4;
  }
  if (!GUARD || col < Nc) {
    float bv = bias ? bias[col] : 0.0f;
#pragma unroll
    for (int r = 0; r < 8; ++r)
      out[(size_t)(r0 + 8 * hi + r) * Nc + col] = acc[r] + bv;
  }
}

// ---------------------------------------------------------------------------
// Apply frames: *_g = R * pts + t    (one block per residue n)
// ---------------------------------------------------------------------------
__global__ void frames_kernel(const float* __restrict__ qp, const float* __restrict__ kp,
                              const float* __restrict__ vp, const float* __restrict__ rot,
                              const float* __restrict__ trans,
                              float* __restrict__ qg, float* __restrict__ kg,
                              float* __restrict__ vg) {
  int n = blockIdx.x;
  const float* R = rot + (size_t)n * 9;
  float tx = trans[n * 3 + 0], ty = trans[n * 3 + 1], tz = trans[n * 3 + 2];
  for (int t = threadIdx.x; t < 2 * HD * PD + HD * PVD; t += blockDim.x) {
    const float* src; float* dst;
    if (t < HD * PD)          { src = qp + (size_t)n * HP3  + t * 3;                         dst = qg + (size_t)n * HP3  + t * 3; }
    else if (t < 2 * HD * PD) { int u = t - HD * PD;   src = kp + (size_t)n * HP3  + u * 3;  dst = kg + (size_t)n * HP3  + u * 3; }
    else                      { int u = t - 2*HD*PD;   src = vp + (size_t)n * HPV3 + u * 3;  dst = vg + (size_t)n * HPV3 + u * 3; }
    float x = src[0], y = src[1], z = src[2];
    dst[0] = R[0]*x + R[1]*y + R[2]*z + tx;
    dst[1] = R[3]*x + R[4]*y + R[5]*z + ty;
    dst[2] = R[6]*x + R[7]*y + R[8]*z + tz;
  }
}

__global__ void sq_kernel(const float* __restrict__ qg, const float* __restrict__ kg,
                          float* __restrict__ qsq, float* __restrict__ ksq) {
  int idx = blockIdx.x * blockDim.x + threadIdx.x;   // n*H + h
  if (idx >= NN * HD) return;
  const float* g = qg + (size_t)idx * (PD * 3);
  float s = 0.f;
#pragma unroll
  for (int i = 0; i < PD * 3; ++i) s += g[i] * g[i];
  qsq[idx] = s;
  g = kg + (size_t)idx * (PD * 3);
  s = 0.f;
#pragma unroll
  for (int i = 0; i < PD * 3; ++i) s += g[i] * g[i];
  ksq[idx] = s;
}

// ---------------------------------------------------------------------------
// Logits: scalar qk (WMMA K=16) + point term (WMMA K=12) + dist bias +
// multiscale + mask, added onto the pair-bias GEMM result pb (n,m,h).
// Output is written HEAD-MAJOR: lattn[h][n][m] (softmax/attn-friendly).
// grid (mTiles, nTiles), block 32.
// ---------------------------------------------------------------------------
__global__ __launch_bounds__(32)
void logits_kernel(const float* __restrict__ q, const float* __restrict__ k,
                   const float* __restrict__ qg, const float* __restrict__ kg,
                   const float* __restrict__ qsq, const float* __restrict__ ksq,
                   const float* __restrict__ trans, const unsigned char* __restrict__ mask,
                   const float* __restrict__ dist_emb, const float* __restrict__ scale_logits,
                   const float* __restrict__ head_weights,
                   const float* __restrict__ pb, float* __restrict__ lattn) {
  int lane = threadIdx.x;
  int lo = lane & 15, hi = lane >> 4;
  int m0 = blockIdx.x * 16, n0 = blockIdx.y * 16;
  int mcol = m0 + lo;

  float tmx = trans[mcol*3], tmy = trans[mcol*3+1], tmz = trans[mcol*3+2];
  float locw[8], medw[8];
  int   bins[8];
#pragma unroll
  for (int r = 0; r < 8; ++r) {
    int n = n0 + 8 * hi + r;
    float dx = trans[n*3] - tmx, dy = trans[n*3+1] - tmy, dz = trans[n*3+2] - tmz;
    float d = sqrtf(dx*dx + dy*dy + dz*dz);
    int b = (int)ceilf(d * 2.0f) - 1;          // searchsorted(linspace(0,32,65))-1
    bins[r] = b < 0 ? 0 : (b > NBD-1 ? NBD-1 : b);
    locw[r] = (d <= 5.0f) ? 1.0f : 0.0f;
    medw[r] = (d > 5.0f && d <= 15.0f) ? 1.0f : 0.0f;
  }
  unsigned char maskm = mask[mcol];

  for (int h = 0; h < HD; ++h) {
    const float* qrow  = q  + (size_t)(n0 + lo) * HC  + h * CD     + 2*hi;
    const float* krow  = k  + (size_t)mcol      * HC  + h * CD     + 2*hi;
    const float* qgrow = qg + (size_t)(n0 + lo) * HP3 + h * (PD*3) + 2*hi;
    const float* kgrow = kg + (size_t)mcol      * HP3 + h * (PD*3) + 2*hi;

    v8f sacc = {};
#pragma unroll
    for (int kk = 0; kk < CD; kk += 4) {
      v2f a = *(const v2f*)(qrow + kk);
      v2f b = *(const v2f*)(krow + kk);
      sacc = wmma4(a, b, sacc);
    }
    v8f pacc = {};
#pragma unroll
    for (int kk = 0; kk < PD*3; kk += 4) {
      v2f a = *(const v2f*)(qgrow + kk);
      v2f b = *(const v2f*)(kgrow + kk);
      pacc = wmma4(a, b, pacc);
    }

    float hw = head_weights[h];
    float e0 = __expf(scale_logits[0*HD + h]);
    float e1 = __expf(scale_logits[1*HD + h]);
    float e2 = __expf(scale_logits[2*HD + h]);
    float inv = 1.0f / (e0 + e1 + e2);
    float ph0 = e0 * inv, ph1 = e1 * inv, ph2 = e2 * inv;
    float kmsq = ksq[mcol * HD + h];

#pragma unroll
    for (int r = 0; r < 8; ++r) {
      int n = n0 + 8 * hi + r;
      float pd = qsq[n * HD + h] + kmsq - 2.0f * pacc[r];
      float l = pb[((size_t)n * NN + mcol) * HD + h]   // pair bias (GEMM)
              + sacc[r] * 0.25f                        // 1/sqrt(C), C=16
              - 0.5f * pd * hw
              + dist_emb[bins[r] * HD + h]
              + (locw[r] * ph0 + medw[r] * ph1 + ph2);
      if (!(mask[n] && maskm)) l = -10000.0f;
      lattn[((size_t)h * NN + n) * NN + mcol] = l;     // head-major
    }
  }
}

// ---------------------------------------------------------------------------
// Softmax over m per (h,n) row, head-major layout, fully coalesced. In place.
// ---------------------------------------------------------------------------
__global__ __launch_bounds__(32)
void softmax_kernel(float* __restrict__ lattn) {
  float* row = lattn + (size_t)blockIdx.x * NN;
  int lane = threadIdx.x;
  float vals[NN / 32];
  float mx = -1e30f;
#pragma unroll
  for (int i = 0; i < NN / 32; ++i) {
    float v = row[lane + 32 * i];
    vals[i] = v;
    mx = fmaxf(mx, v);
  }
  for (int o = 16; o > 0; o >>= 1) mx = fmaxf(mx, __shfl_xor(mx, o, 32));
  float sum = 0.f;
#pragma unroll
  for (int i = 0; i < NN / 32; ++i) { vals[i] = __expf(vals[i] - mx); sum += vals[i]; }
  for (int o = 16; o > 0; o >>= 1) sum += __shfl_xor(sum, o, 32);
  float inv = 1.0f / sum;
#pragma unroll
  for (int i = 0; i < NN / 32; ++i) row[lane + 32 * i] = vals[i] * inv;
}

// ---------------------------------------------------------------------------
// out_scalar[n,h,c] = sum_m attn[h][n][m] vT[h*16+c][m]   (grid: nTiles x H)
// Both fragments are contiguous b64 loads.
// ---------------------------------------------------------------------------
__global__ __launch_bounds__(32)
void attn_v_kernel(const float* __restrict__ lattn, const float* __restrict__ vT,
                   float* __restrict__ out) {
  int lane = threadIdx.x, lo = lane & 15, hi = lane >> 4;
  int n0 = blockIdx.x * 16, h = blockIdx.y;
  const float* Ap = lattn + ((size_t)h * NN + n0 + lo) * NN + 2 * hi;
  const float* Bp = vT + (size_t)(h * CD + lo) * NN + 2 * hi;
  v8f acc = {};
#pragma unroll 4
  for (int kk = 0; kk < NN; kk += 4) {
    v2f a = *(const v2f*)Ap;
    v2f b = *(const v2f*)Bp;
    acc = wmma4(a, b, acc);
    Ap += 4; Bp += 4;
  }
#pragma unroll
  for (int r = 0; r < 8; ++r)
    out[(size_t)(n0 + 8*hi + r) * HC + h * CD + lo] = acc[r];
}

// out_pts_g[n,h,d] (d<24) = sum_m attn[h][n][m] vgT[h*24+d][m] (grid nTiles,H,2)
// Cols 24..31 read clamped row 23 (garbage -> unstored D cols); no masking.
__global__ __launch_bounds__(32)
void attn_vg_kernel(const float* __restrict__ lattn, const float* __restrict__ vgT,
                    float* __restrict__ outg) {
  int lane = threadIdx.x, lo = lane & 15, hi = lane >> 4;
  int n0 = blockIdx.x * 16, h = blockIdx.y, ct = blockIdx.z;
  int col = ct * 16 + lo;
  int colc = col < PVD*3 ? col : PVD*3 - 1;
  const float* Ap = lattn + ((size_t)h * NN + n0 + lo) * NN + 2 * hi;
  const float* Bp = vgT + (size_t)(h * PVD*3 + colc) * NN + 2 * hi;
  v8f acc = {};
#pragma unroll 4
  for (int kk = 0; kk < NN; kk += 4) {
    v2f a = *(const v2f*)Ap;
    v2f b = *(const v2f*)Bp;
    acc = wmma4(a, b, acc);
    Ap += 4; Bp += 4;
  }
  if (col < PVD*3)
#pragma unroll
    for (int r = 0; r < 8; ++r)
      outg[(size_t)(n0 + 8*hi + r) * HPV3 + h * (PVD*3) + col] = acc[r];
}

// pair_feat[n,h,c] = sum_m attn[h][n][m] z[n,m,c]; heads are the M dim.
// A rows 12..15 read clamped head 11 (garbage -> unstored D rows).
// grid: n x (CZ/16)
__global__ __launch_bounds__(32)
void pair_feat_kernel(const float* __restrict__ lattn, const float* __restrict__ z,
                      float* __restrict__ pf) {
  int lane = threadIdx.x, lo = lane & 15, hi = lane >> 4;
  int n = blockIdx.x, c0 = blockIdx.y * 16;
  int hc = lo < HD ? lo : HD - 1;
  const float* Ap = lattn + ((size_t)hc * NN + n) * NN + 2 * hi;
  const float* Zp = z + ((size_t)n * NN + 2 * hi) * CZD + c0 + lo;
  v8f acc = {};
#pragma unroll 4
  for (int kk = 0; kk < NN; kk += 4) {
    __builtin_prefetch(Zp + 16 * CZD, 0, 0);   // stream z toward L2
    v2f a = *(const v2f*)Ap;
    v2f b;
    b[0] = Zp[0];
    b[1] = Zp[CZD];
    acc = wmma4(a, b, acc);
    Ap += 4; Zp += 4 * CZD;
  }
#pragma unroll
  for (int r = 0; r < 8; ++r) {
    int hrow = 8 * hi + r;
    if (hrow < HD)
      pf[((size_t)n * HD + hrow) * CZD + c0 + lo] = acc[r];
  }
}

// ---------------------------------------------------------------------------
// Assemble feats[n, h*176 + {scalar16 | pts_l24 | norms8 | pair128}]
// ---------------------------------------------------------------------------
__global__ void assemble_kernel(const float* __restrict__ osc, const float* __restrict__ opg,
                                const float* __restrict__ pf, const float* __restrict__ rot,
                                const float* __restrict__ trans, float* __restrict__ feats) {
  int n = blockIdx.x;
  const float* R = rot + (size_t)n * 9;
  float tx = trans[n*3], ty = trans[n*3+1], tz = trans[n*3+2];
  const int per = CD + PVD*3 + PVD + CZD;   // 176
  for (int idx = threadIdx.x; idx < HD * per; idx += blockDim.x) {
    int h = idx / per, o = idx % per;
    float val;
    if (o < CD) {
      val = osc[(size_t)n * HC + h * CD + o];
    } else if (o < CD + PVD*3) {
      int d = o - CD, p = d / 3, i = d % 3;
      const float* g = opg + (size_t)n * HPV3 + h * (PVD*3) + p * 3;
      float gx = g[0]-tx, gy = g[1]-ty, gz = g[2]-tz;
      val = R[0*3+i]*gx + R[1*3+i]*gy + R[2*3+i]*gz;  // R^T * (g - t)
    } else if (o < CD + PVD*4) {
      int p = o - (CD + PVD*3);
      const float* g = opg + (size_t)n * HPV3 + h * (PVD*3) + p * 3;
      float gx = g[0]-tx, gy = g[1]-ty, gz = g[2]-tz;
      float v0 = R[0]*gx + R[3]*gy + R[6]*gz;
      float v1 = R[1]*gx + R[4]*gy + R[7]*gz;
      float v2 = R[2]*gx + R[5]*gy + R[8]*gz;
      val = sqrtf(v0*v0 + v1*v1 + v2*v2);
    } else {
      val = pf[((size_t)n * HD + h) * CZD + (o - (CD + PVD*4))];
    }
    feats[(size_t)n * FEAT + idx] = val;
  }
}

// ---------------------------------------------------------------------------
extern "C" void kernel_launch(void* const* d_in, const int* in_sizes, int n_in,
                              void* d_out, int out_size, void* d_ws, size_t ws_size,
                              hipStream_t stream) {
  (void)in_sizes; (void)n_in; (void)out_size; (void)ws_size;
  const float* s     = (const float*)d_in[0];
  const float* z     = (const float*)d_in[1];
  const float* trans = (const float*)d_in[2];
  const float* rot   = (const float*)d_in[3];
  const unsigned char* mask = (const unsigned char*)d_in[4];
  const float* Wq  = (const float*)d_in[5],  *bq  = (const float*)d_in[6];
  const float* Wk  = (const float*)d_in[7],  *bk  = (const float*)d_in[8];
  const float* Wv  = (const float*)d_in[9],  *bv  = (const float*)d_in[10];
  const float* Wqp = (const float*)d_in[11], *bqp = (const float*)d_in[12];
  const float* Wkp = (const float*)d_in[13], *bkp = (const float*)d_in[14];
  const float* Wvp = (const float*)d_in[15], *bvp = (const float*)d_in[16];
  const float* Wb  = (const float*)d_in[17], *bb  = (const float*)d_in[18];
  const float* dist_emb     = (const float*)d_in[19];
  const float* scale_logits = (const float*)d_in[20];
  const float* head_weights = (const float*)d_in[21];
  const float* Wout = (const float*)d_in[22], *bout = (const float*)d_in[23];

  float* ws = (float*)d_ws;
  // Transposed / padded weights
  float* WqT   = ws; ws += (size_t)HC   * CSD;
  float* WkT   = ws; ws += (size_t)HC   * CSD;
  float* WvT   = ws; ws += (size_t)HC   * CSD;
  float* WqpT  = ws; ws += (size_t)HP3  * CSD;
  float* WkpT  = ws; ws += (size_t)HP3  * CSD;
  float* WvpT  = ws; ws += (size_t)HPV3 * CSD;
  float* WbT   = ws; ws += (size_t)16   * CZD;    // padded 12 -> 16
  float* WoutT = ws; ws += (size_t)CSD  * FEAT;
  // Activations
  float* q     = ws; ws += (size_t)NN * HC;
  float* k     = ws; ws += (size_t)NN * HC;
  float* v     = ws; ws += (size_t)NN * HC;
  float* qp    = ws; ws += (size_t)NN * HP3;
  float* kp    = ws; ws += (size_t)NN * HP3;
  float* vp    = ws; ws += (size_t)NN * HPV3;
  float* qg    = ws; ws += (size_t)NN * HP3;
  float* kg    = ws; ws += (size_t)NN * HP3;
  float* vg    = ws; ws += (size_t)NN * HPV3;
  float* vT    = ws; ws += (size_t)NN * HC;       // [h*16+c][m]
  float* vgT   = ws; ws += (size_t)NN * HPV3;     // [h*24+d][m]
  float* qsq   = ws; ws += (size_t)NN * HD;
  float* ksq   = ws; ws += (size_t)NN * HD;
  float* pb    = ws; ws += (size_t)NN * NN * HD;  // pair bias (n,m,h)
  float* lattn = ws; ws += (size_t)NN * NN * HD;  // logits/attn [h][n][m]
  float* osc   = ws; ws += (size_t)NN * HC;
  float* opg   = ws; ws += (size_t)NN * HPV3;
  float* pf    = ws; ws += (size_t)NN * HD * CZD;
  float* feats = ws; ws += (size_t)NN * FEAT;

  const int TB = 256;
  auto tgrid = [](size_t n) { return (unsigned)((n + 255) / 256); };

  // Weight transposes (+ pad Wb to 16 cols)
  transpose_pad_kernel<<<tgrid((size_t)HC*CSD),   TB, 0, stream>>>(Wq,   WqT,   CSD, HC,   HC);
  transpose_pad_kernel<<<tgrid((size_t)HC*CSD),   TB, 0, stream>>>(Wk,   WkT,   CSD, HC,   HC);
  transpose_pad_kernel<<<tgrid((size_t)HC*CSD),   TB, 0, stream>>>(Wv,   WvT,   CSD, HC,   HC);
  transpose_pad_kernel<<<tgrid((size_t)HP3*CSD),  TB, 0, stream>>>(Wqp,  WqpT,  CSD, HP3,  HP3);
  transpose_pad_kernel<<<tgrid((size_t)HP3*CSD),  TB, 0, stream>>>(Wkp,  WkpT,  CSD, HP3,  HP3);
  transpose_pad_kernel<<<tgrid((size_t)HPV3*CSD), TB, 0, stream>>>(Wvp,  WvpT,  CSD, HPV3, HPV3);
  transpose_pad_kernel<<<tgrid((size_t)16*CZD),   TB, 0, stream>>>(Wb,   WbT,   CZD, HD,   16);
  transpose_pad_kernel<<<tgrid((size_t)CSD*FEAT), TB, 0, stream>>>(Wout, WoutT, FEAT, CSD, CSD);

  // Projections: s(512x384) @ W + b
  gemm_bt_kernel<false><<<dim3(NN/16, HC/16),   32, 0, stream>>>(s, WqT,  bq,  q,  CSD, HC);
  gemm_bt_kernel<false><<<dim3(NN/16, HC/16),   32, 0, stream>>>(s, WkT,  bk,  k,  CSD, HC);
  gemm_bt_kernel<false><<<dim3(NN/16, HC/16),   32, 0, stream>>>(s, WvT,  bv,  v,  CSD, HC);
  gemm_bt_kernel<false><<<dim3(NN/16, HP3/16),  32, 0, stream>>>(s, WqpT, bqp, qp, CSD, HP3);
  gemm_bt_kernel<false><<<dim3(NN/16, HP3/16),  32, 0, stream>>>(s, WkpT, bkp, kp, CSD, HP3);
  gemm_bt_kernel<false><<<dim3(NN/16, HPV3/16), 32, 0, stream>>>(s, WvpT, bvp, vp, CSD, HPV3);

  // Frames, norms, per-head transposes of v / v_g
  frames_kernel<<<NN, 192, 0, stream>>>(qp, kp, vp, rot, trans, qg, kg, vg);
  sq_kernel<<<(NN*HD + 255)/256, 256, 0, stream>>>(qg, kg, qsq, ksq);
  transpose_pad_kernel<<<tgrid((size_t)NN*HC),   TB, 0, stream>>>(v,  vT,  NN, HC,   HC);
  transpose_pad_kernel<<<tgrid((size_t)NN*HPV3), TB, 0, stream>>>(vg, vgT, NN, HPV3, HPV3);

  // Pair bias: z flattened (262144 x 128) @ WbT(16pad x 128) -> pb (n,m,h)
  gemm_bt_kernel<true><<<dim3((NN*NN)/16, 1), 32, 0, stream>>>(z, WbT, bb, pb, CZD, HD);

  // Remaining logit terms + mask -> head-major lattn, then softmax in place
  logits_kernel<<<dim3(NN/16, NN/16), 32, 0, stream>>>(q, k, qg, kg, qsq, ksq,
                                                       trans, mask, dist_emb,
                                                       scale_logits, head_weights,
                                                       pb, lattn);
  softmax_kernel<<<HD*NN, 32, 0, stream>>>(lattn);

  // Attention outputs
  attn_v_kernel<<<dim3(NN/16, HD),     32, 0, stream>>>(lattn, vT,  osc);
  attn_vg_kernel<<<dim3(NN/16, HD, 2), 32, 0, stream>>>(lattn, vgT, opg);
  pair_feat_kernel<<<dim3(NN, CZD/16), 32, 0, stream>>>(lattn, z, pf);

  // Assemble feats and final projection
  assemble_kernel<<<NN, 256, 0, stream>>>(osc, opg, pf, rot, trans, feats);
  gemm_bt_kernel<false><<<dim3(NN/16, CSD/16), 32, 0, stream>>>(feats, WoutT, bout,
                                                                (float*)d_out,
                                                                FEAT, CSD);
}